// RecognitionModel_26276609917187
// MI455X (gfx1250) — compile-verified
//
#include <hip/hip_runtime.h>
#include <hip/hip_bf16.h>
#include <math.h>
#include <stdint.h>

// Problem constants (from reference): B=XD=64, NTRIALS=16, T=512, NMC=32
#define NT     16
#define T_LEN  512
#define NMC    32
#define JITF   1e-6f

typedef float v2f __attribute__((ext_vector_type(2)));
typedef float v8f __attribute__((ext_vector_type(8)));
typedef unsigned int u32x4 __attribute__((ext_vector_type(4)));
typedef int i32x4 __attribute__((ext_vector_type(4)));
typedef int i32x8 __attribute__((ext_vector_type(8)));

// LDS (addrspace 3) pointer: 32-bit addressing, no generic null-check lowering.
typedef __attribute__((address_space(3))) float ldsf;
__device__ inline ldsf* to_lds(float* p) {
  return (ldsf*)(unsigned int)(uintptr_t)p;   // aperture: LDS offset = addr[31:0]
}

#define V8F_ZERO {0.f,0.f,0.f,0.f,0.f,0.f,0.f,0.f}

// ---------------------------------------------------------------------------
// Tiled fp32 WMMA GEMM: D(MxN) = op(A)(MxK) * op(B)(KxN). M,N mult of 16,
// K mult of 8. 8 waves, dual accumulators to break the WMMA RAW chain.
// Pointer types templated: float* (global) or ldsf* (LDS, 32-bit addressing).
// ---------------------------------------------------------------------------
template <bool TA, bool TB, typename PD, typename PA, typename PB>
__device__ inline void gemm_tiles(PD D, int ldd, PA A, int lda, PB B, int ldb,
                                  int M, int N, int K) {
  const int tid  = threadIdx.x;
  const int wave = tid >> 5;
  const int lane = tid & 31;
  const int half = lane >> 4;
  const int l15  = lane & 15;
  const int tn   = N >> 4;
  const int ntiles = (M >> 4) * tn;
  for (int tile = wave; tile < ntiles; tile += 8) {
    const int mi = (tile / tn) * 16;
    const int ni = (tile % tn) * 16;
    v8f acc0 = V8F_ZERO, acc1 = V8F_ZERO;
    for (int k0 = 0; k0 < K; k0 += 8) {
#pragma unroll
      for (int u = 0; u < 2; ++u) {
        const int k = k0 + u * 4 + half * 2;
        v2f av, bv;
        if (TA) { av.x = A[k * lda + mi + l15];       av.y = A[(k + 1) * lda + mi + l15]; }
        else    { av.x = A[(mi + l15) * lda + k];     av.y = A[(mi + l15) * lda + k + 1]; }
        if (TB) { bv.x = B[(ni + l15) * ldb + k];     bv.y = B[(ni + l15) * ldb + k + 1]; }
        else    { bv.x = B[k * ldb + ni + l15];       bv.y = B[(k + 1) * ldb + ni + l15]; }
        if (u == 0)
          acc0 = __builtin_amdgcn_wmma_f32_16x16x4_f32(false, av, false, bv, (short)0, acc0, false, false);
        else
          acc1 = __builtin_amdgcn_wmma_f32_16x16x4_f32(false, av, false, bv, (short)0, acc1, false, false);
      }
    }
    const v8f acc = acc0 + acc1;
#pragma unroll
    for (int j = 0; j < 8; ++j)
      D[(mi + half * 8 + j) * ldd + (ni + l15)] = acc[j];
  }
}

// ---------------------------------------------------------------------------
// Blocked Cholesky of 64x64 in LDS (lower factor; upper stale).
// 16x16 diagonal panels scalar, rank-16 trailing updates via WMMA.
// ---------------------------------------------------------------------------
__device__ inline void chol64_blocked(ldsf* S, ldsf* scratch) {
  const int tid = threadIdx.x;
  for (int kb = 0; kb < 64; kb += 16) {
    for (int k = kb; k < kb + 16; ++k) {
      __syncthreads();
      if (tid == 0) S[k * 64 + k] = sqrtf(S[k * 64 + k]);
      __syncthreads();
      const float inv = 1.0f / S[k * 64 + k];
      for (int i = k + 1 + tid; i < kb + 16; i += 256) S[i * 64 + k] *= inv;
      __syncthreads();
      const int rem = kb + 15 - k;
      for (int idx = tid; idx < rem * rem; idx += 256) {
        const int i = k + 1 + idx / rem;
        const int j = k + 1 + idx % rem;
        if (j <= i) S[i * 64 + j] -= S[i * 64 + k] * S[j * 64 + k];
      }
    }
    __syncthreads();
    const int nrows = 64 - (kb + 16);
    if (nrows > 0) {
      if (tid < nrows) {                       // panel solve vs Ldd^T
        const int r = kb + 16 + tid;
        for (int c = 0; c < 16; ++c) {
          float acc = S[r * 64 + kb + c];
          for (int j = 0; j < c; ++j)
            acc -= S[r * 64 + kb + j] * S[(kb + c) * 64 + kb + j];
          S[r * 64 + kb + c] = acc / S[(kb + c) * 64 + kb + c];
        }
      }
      __syncthreads();
      // trailing update via WMMA: S22 -= P * P^T
      gemm_tiles<false, true>(scratch, 64, S + (kb + 16) * 64 + kb, 64,
                              S + (kb + 16) * 64 + kb, 64, nrows, nrows, 16);
      __syncthreads();
      for (int idx = tid; idx < nrows * nrows; idx += 256) {
        const int i = idx / nrows, j = idx % nrows;
        S[(kb + 16 + i) * 64 + kb + 16 + j] -= scratch[i * 64 + j];
      }
      __syncthreads();
    }
  }
  __syncthreads();
}

// ---------------------------------------------------------------------------
// Explicit inverse of lower-triangular 64x64: V = L^{-1} (lower, upper = 0).
// Diagonal 16x16 blocks: 64 independent column solves. Off-diagonal blocks:
// V_ij = -Dinv_i * (sum_k L_ik V_kj) via per-wave 16x16 WMMA GEMMs.
// ---------------------------------------------------------------------------
__device__ inline void trinv64(const ldsf* L, ldsf* V, ldsf* scratch) {
  const int tid  = threadIdx.x;
  const int wave = tid >> 5;
  const int lane = tid & 31;
  const int half = lane >> 4;
  const int l15  = lane & 15;
  for (int idx = tid; idx < 4096; idx += 256) V[idx] = 0.f;
  __syncthreads();
  if (tid < 64) {                       // 4 diag blocks x 16 columns
    const int b = tid >> 4, c = tid & 15, o = b * 16;
    float x[16];
#pragma unroll
    for (int i = 0; i < 16; ++i) {
      float acc = (i == c) ? 1.f : 0.f;
#pragma unroll
      for (int j = 0; j < 16; ++j)
        if (j < i) acc -= L[(o + i) * 64 + o + j] * x[j];
      x[i] = acc / L[(o + i) * 64 + o + i];
    }
#pragma unroll
    for (int i = 0; i < 16; ++i) V[(o + i) * 64 + o + c] = x[i];
  }
  __syncthreads();
  for (int d = 1; d <= 3; ++d) {
    const int npairs = 4 - d;
    if (wave < npairs) {                // stage 1: T_p = L_i,[j..i) * V_[j..i),j
      const int j = wave, i = j + d;
      v8f acc = V8F_ZERO;
      for (int k0 = 0; k0 < 16 * d; k0 += 4) {
        const int k = k0 + half * 2;
        v2f av, bv;
        av.x = L[(i * 16 + l15) * 64 + j * 16 + k];
        av.y = L[(i * 16 + l15) * 64 + j * 16 + k + 1];
        bv.x = V[(j * 16 + k) * 64 + j * 16 + l15];
        bv.y = V[(j * 16 + k + 1) * 64 + j * 16 + l15];
        acc = __builtin_amdgcn_wmma_f32_16x16x4_f32(false, av, false, bv, (short)0, acc, false, false);
      }
#pragma unroll
      for (int r = 0; r < 8; ++r)
        scratch[wave * 256 + (half * 8 + r) * 16 + l15] = acc[r];
    }
    __syncthreads();
    if (wave < npairs) {                // stage 2: V_ij = -Dinv_i * T_p
      const int j = wave, i = j + d;
      v8f acc = V8F_ZERO;
      for (int k0 = 0; k0 < 16; k0 += 4) {
        const int k = k0 + half * 2;
        v2f av, bv;
        av.x = V[(i * 16 + l15) * 64 + i * 16 + k];
        av.y = V[(i * 16 + l15) * 64 + i * 16 + k + 1];
        bv.x = scratch[wave * 256 + k * 16 + l15];
        bv.y = scratch[wave * 256 + (k + 1) * 16 + l15];
        acc = __builtin_amdgcn_wmma_f32_16x16x4_f32(false, av, false, bv, (short)0, acc, false, false);
      }
#pragma unroll
      for (int r = 0; r < 8; ++r)
        V[(i * 16 + half * 8 + r) * 64 + j * 16 + l15] = -acc[r];
    }
    __syncthreads();
  }
}

__device__ inline void sym64_inplace(ldsf* M) {
  const int tid = threadIdx.x;
  __syncthreads();
  for (int idx = tid; idx < 4096; idx += 256) {
    const int i = idx >> 6, j = idx & 63;
    if (j < i) {
      const float v = 0.5f * (M[i * 64 + j] + M[j * 64 + i]);
      M[i * 64 + j] = v;
      M[j * 64 + i] = v;
    }
  }
  __syncthreads();
}

__device__ inline void transpose64_inplace(ldsf* M) {
  __syncthreads();
  for (int idx = threadIdx.x; idx < 4096; idx += 256) {
    const int i = idx >> 6, j = idx & 63;
    if (j < i) {
      const float a = M[i * 64 + j], b = M[j * 64 + i];
      M[i * 64 + j] = b;
      M[j * 64 + i] = a;
    }
  }
  __syncthreads();
}

template <typename PD, typename PS>
__device__ inline void copy64(PD d, PS s) {
  for (int idx = threadIdx.x; idx < 4096; idx += 256) d[idx] = s[idx];
}

// ---------------------------------------------------------------------------
// Tensor Data Mover: DMA a 64x64 fp32 tile (global, row stride 64) into LDS.
// 6-arg builtin form (clang-23 / therock-10.0). Issued by wave 0; caller must
// __syncthreads() after.
// ---------------------------------------------------------------------------
#if defined(__has_builtin)
#if __has_builtin(__builtin_amdgcn_tensor_load_to_lds) && __has_builtin(__builtin_amdgcn_s_wait_tensorcnt)
#define HAVE_TDM 1
#endif
#endif

__device__ inline void tdm_load_64x64(float* lds_dst, const float* gsrc) {
#ifdef HAVE_TDM
  if (threadIdx.x < 32) {
    const unsigned long long ga = (unsigned long long)(uintptr_t)gsrc;
    const unsigned int la = (unsigned int)(uintptr_t)lds_dst;  // LDS offset
    u32x4 g0;
    g0[0] = 1u;                                                // count=1
    g0[1] = la;                                                // lds_addr
    g0[2] = (unsigned int)ga;                                  // global_addr[31:0]
    g0[3] = (unsigned int)((ga >> 32) & 0x1FFFFFFu) | (2u << 30);  // addr[56:32] | type=2
    i32x8 g1;
    g1[0] = (int)(2u << 16);        // data_size=4B
    g1[1] = (int)(64u << 16);       // tensor_dim0 = 64
    g1[2] = (int)(64u << 16);       // tensor_dim1 = 64
    g1[3] = (int)(64u << 16);       // tile_dim0 = 64
    g1[4] = (int)64;                // tile_dim1 = 64, tile_dim2 = 0
    g1[5] = (int)64;                // tensor_dim0_stride = 64
    g1[6] = (int)(4096u << 16);     // tensor_dim1_stride = 4096
    g1[7] = 0;
    const i32x4 z4 = {0, 0, 0, 0};
    const i32x8 z8 = {0, 0, 0, 0, 0, 0, 0, 0};
    __builtin_amdgcn_tensor_load_to_lds(g0, g1, z4, z4, z8, 0);
    __builtin_amdgcn_s_wait_tensorcnt(0);
  }
#else
  copy64(to_lds(lds_dst), gsrc);
#endif
}

// ---------------------------------------------------------------------------
// Kernel 0: one-shot transpose of x_hat [NMC,NT,64,T] -> xt [T,NMC,NT,64]
// ---------------------------------------------------------------------------
__global__ void __launch_bounds__(256)
xhat_transpose(const float* __restrict__ x_hat, float* __restrict__ xt) {
  const size_t total = (size_t)T_LEN * NMC * NT * 64;
  for (size_t o = (size_t)blockIdx.x * 256 + threadIdx.x; o < total;
       o += (size_t)gridDim.x * 256) {
    const int i = (int)(o & 63);
    const int trial = (int)((o >> 6) & 15);
    const int c = (int)((o >> 10) & 31);
    const int t = (int)(o >> 15);
    xt[o] = x_hat[(((size_t)c * NT + trial) * 64 + i) * T_LEN + t];
  }
}

// ---------------------------------------------------------------------------
// Kernel 1: covariance filter + fused means filter. One block per trial.
// ---------------------------------------------------------------------------
__global__ void __launch_bounds__(256)
kf_filter(const float* __restrict__ A, const float* __restrict__ W,
          const float* __restrict__ Q, const float* __restrict__ R,
          const float* __restrict__ Sigma0, const float* __restrict__ mu0,
          const float* __restrict__ xt,
          float* __restrict__ Sf_hist, float* __restrict__ Sd_hist,
          float* __restrict__ L_hist, float* __restrict__ md_hist,
          float* __restrict__ mus_filt) {
  __shared__ float sSf_[4096], sSd_[4096], sT1_[4096], sM_[4096], sS_[4096], sK_[4096];
  __shared__ float sMf_[2048], sMd_[2048], sU_[2048], sV_[2048];   // 64x32, ld=32
  ldsf* sSf = to_lds(sSf_);  ldsf* sSd = to_lds(sSd_);  ldsf* sT1 = to_lds(sT1_);
  ldsf* sM  = to_lds(sM_);   ldsf* sS  = to_lds(sS_);   ldsf* sK  = to_lds(sK_);
  ldsf* sMf = to_lds(sMf_);  ldsf* sMd = to_lds(sMd_);
  ldsf* sU  = to_lds(sU_);   ldsf* sV  = to_lds(sV_);
  const int trial = blockIdx.x;
  const int tid = threadIdx.x;

  for (int t = 0; t < T_LEN; ++t) {
    if (t == 0) {
      copy64(sSd, Sigma0 + (size_t)trial * 4096);
      __syncthreads();
    } else {
      gemm_tiles<false, false>(sT1, 64, A, 64, sSf, 64, 64, 64, 64);  // T1 = A*Sf
      __syncthreads();
      gemm_tiles<false, true>(sSd, 64, sT1, 64, A, 64, 64, 64, 64);   // Sd = T1*A^T
      __syncthreads();
      for (int idx = tid; idx < 4096; idx += 256) sSd[idx] += Q[idx];
      sym64_inplace(sSd);
      {
        float* g = Sd_hist + ((size_t)(t - 1) * NT + trial) * 4096;
        for (int idx = tid; idx < 4096; idx += 256) g[idx] = sSd[idx];
      }
      copy64(sT1, sSd);
      __syncthreads();
      if (tid < 64) sT1[tid * 64 + tid] += JITF;
      chol64_blocked(sT1, sK);                       // L = chol(Sd + JIT I)
      {
        float* g = L_hist + ((size_t)(t - 1) * NT + trial) * 4096;
        for (int idx = tid; idx < 4096; idx += 256) g[idx] = sT1[idx];
      }
      __syncthreads();
    }
    // innovation covariance and gain
    gemm_tiles<false, false>(sM, 64, W, 64, sSd, 64, 64, 64, 64);    // M = W*Sd
    __syncthreads();
    gemm_tiles<false, true>(sS, 64, sM, 64, W, 64, 64, 64, 64);      // S = M*W^T
    __syncthreads();
    for (int idx = tid; idx < 4096; idx += 256) sS[idx] += R[idx];
    if (t > 0 && tid < 64) sS[tid * 64 + tid] += JITF;
    chol64_blocked(sS, sK);                                          // S = chol(.)
    trinv64(sS, sT1, sK);                                            // T1 = S^{-1}
    __syncthreads();
    gemm_tiles<false, false>(sS, 64, sT1, 64, sM, 64, 64, 64, 64);   // Y = Sinv*M
    __syncthreads();
    gemm_tiles<true, false>(sK, 64, sT1, 64, sS, 64, 64, 64, 64);    // Z = Sinv^T*Y
    transpose64_inplace(sK);                                         // K = Z^T
    gemm_tiles<false, false>(sT1, 64, sK, 64, sM, 64, 64, 64, 64);   // T1 = K*M
    __syncthreads();
    for (int idx = tid; idx < 4096; idx += 256) sSf[idx] = sSd[idx] - sT1[idx];
    sym64_inplace(sSf);                                              // Sf
    {
      float* g = Sf_hist + ((size_t)t * NT + trial) * 4096;
      for (int idx = tid; idx < 4096; idx += 256) g[idx] = sSf[idx];
    }
    // ---- fused means filter ----
    if (t == 0) {
      for (int idx = tid; idx < 2048; idx += 256)
        sMd[idx] = mu0[trial * 64 + (idx >> 5)];
      __syncthreads();
    } else {
      gemm_tiles<false, false>(sMd, 32, A, 64, sMf, 32, 64, 32, 64); // md = A*mf
      __syncthreads();
      float* g = md_hist + ((size_t)(t - 1) * NMC * NT) * 64 + trial * 64;
      for (int idx = tid; idx < 2048; idx += 256) {
        const int i = idx >> 5, c = idx & 31;
        g[(size_t)c * NT * 64 + i] = sMd[idx];
      }
    }
    {                                             // coalesced x_hat[t] load
      const float* g = xt + ((size_t)t * NMC * NT) * 64 + trial * 64;
      for (int idx = tid; idx < 2048; idx += 256) {
        const int c = idx >> 6, i = idx & 63;
        sV[i * 32 + c] = g[(size_t)c * NT * 64 + i];
      }
    }
    __syncthreads();
    gemm_tiles<false, false>(sU, 32, W, 64, sMd, 32, 64, 32, 64);    // U = W*md
    __syncthreads();
    for (int idx = tid; idx < 2048; idx += 256) sU[idx] = sV[idx] - sU[idx];
    __syncthreads();
    gemm_tiles<false, false>(sV, 32, sK, 64, sU, 32, 64, 32, 64);    // V = K*innov
    __syncthreads();
    {
      float* g = mus_filt + ((size_t)t * NMC * NT) * 64 + trial * 64;
      for (int idx = tid; idx < 2048; idx += 256) {
        const int i = idx >> 5, c = idx & 31;
        const float v = sMd[idx] + sV[idx];
        sMf[idx] = v;
        g[(size_t)c * NT * 64 + i] = v;
      }
    }
    __syncthreads();
  }
}

// ---------------------------------------------------------------------------
// Kernel 2: smoother covariances + output Cholesky. Parallel over T*NT.
// ---------------------------------------------------------------------------
__global__ void __launch_bounds__(256)
kf_smooth_cov(const float* __restrict__ A, const float* __restrict__ Sf_hist,
              const float* __restrict__ Sd_hist, const float* __restrict__ L_hist,
              float* __restrict__ Cs_ws, float* __restrict__ out_chol) {
  __shared__ float sSf_[4096], sP_[4096], sL_[4096], sCs_[4096], sSd_[4096], sT_[4096];
  ldsf* sSf = to_lds(sSf_);  ldsf* sP = to_lds(sP_);   ldsf* sL = to_lds(sL_);
  ldsf* sCs = to_lds(sCs_);  ldsf* sSd = to_lds(sSd_); ldsf* sT = to_lds(sT_);
  const int t = blockIdx.x / NT;
  const int trial = blockIdx.x % NT;
  const int tid = threadIdx.x;

  tdm_load_64x64(sSf_, Sf_hist + ((size_t)t * NT + trial) * 4096);
  if (t < T_LEN - 1) {
    tdm_load_64x64(sL_, L_hist + ((size_t)t * NT + trial) * 4096);
    tdm_load_64x64(sSd_, Sd_hist + ((size_t)t * NT + trial) * 4096);
    __syncthreads();
    trinv64(sL, sT, sP);                                            // T = L^{-1}
    __syncthreads();
    gemm_tiles<false, false>(sP, 64, A, 64, sSf, 64, 64, 64, 64);   // P = A*Sf
    __syncthreads();
    gemm_tiles<false, false>(sL, 64, sT, 64, sP, 64, 64, 64, 64);   // Y = Linv*P
    __syncthreads();
    gemm_tiles<true, false>(sP, 64, sT, 64, sL, 64, 64, 64, 64);    // Z = Linv^T*Y
    __syncthreads();
    for (int idx = tid; idx < 4096; idx += 256) {                   // Cs = Z^T
      const int i = idx >> 6, j = idx & 63;
      sCs[i * 64 + j] = sP[j * 64 + i];
    }
    __syncthreads();
    {
      float* g = Cs_ws + ((size_t)t * NT + trial) * 4096;
      for (int idx = tid; idx < 4096; idx += 256) g[idx] = sCs[idx];
    }
    gemm_tiles<false, false>(sP, 64, sCs, 64, sSd, 64, 64, 64, 64); // P = Cs*Sd
    __syncthreads();
    gemm_tiles<false, true>(sL, 64, sP, 64, sCs, 64, 64, 64, 64);   // L = P*Cs^T
    __syncthreads();
    for (int idx = tid; idx < 4096; idx += 256) sSf[idx] -= sL[idx];
    sym64_inplace(sSf);                                             // St
  } else {
    __syncthreads();
  }
  if (tid < 64) sSf[tid * 64 + tid] += 1e-4f;
  chol64_blocked(sSf, sP);
  float* g = out_chol + ((size_t)t * NT + trial) * 4096;
  for (int idx = tid; idx < 4096; idx += 256) {
    const int i = idx >> 6, j = idx & 63;
    g[idx] = (j <= i) ? sSf[idx] : 0.0f;      // lower-tri like jnp.linalg.cholesky
  }
}

// ---------------------------------------------------------------------------
// Kernel 3: backward means smoother. One block per trial, persistent loop.
// ---------------------------------------------------------------------------
__global__ void __launch_bounds__(256)
kf_smooth_mean(const float* __restrict__ Cs_ws, const float* __restrict__ md_hist,
               const float* __restrict__ mus_filt, float* __restrict__ out_mu) {
  __shared__ float sC_[4096];
  __shared__ float sMu_[2048], sTmp_[2048], sRes_[2048];
  ldsf* sC = to_lds(sC_);   ldsf* sMu = to_lds(sMu_);
  ldsf* sTmp = to_lds(sTmp_); ldsf* sRes = to_lds(sRes_);
  const int trial = blockIdx.x;
  const int tid = threadIdx.x;

  for (int idx = tid; idx < 2048; idx += 256) {
    const int i = idx >> 5, c = idx & 31;
    const size_t off = (((size_t)(T_LEN - 1) * NMC + c) * NT + trial) * 64 + i;
    const float v = mus_filt[off];
    sMu[idx] = v;
    out_mu[off] = v;
  }
  __syncthreads();
  for (int t = T_LEN - 2; t >= 0; --t) {
    tdm_load_64x64(sC_, Cs_ws + ((size_t)t * NT + trial) * 4096);
    for (int idx = tid; idx < 2048; idx += 256) {
      const int i = idx >> 5, c = idx & 31;
      sTmp[idx] = sMu[idx] - md_hist[(((size_t)t * NMC + c) * NT + trial) * 64 + i];
    }
    __syncthreads();
    gemm_tiles<false, false>(sRes, 32, sC, 64, sTmp, 32, 64, 32, 64);
    __syncthreads();
    for (int idx = tid; idx < 2048; idx += 256) {
      const int i = idx >> 5, c = idx & 31;
      const size_t off = (((size_t)t * NMC + c) * NT + trial) * 64 + i;
      const float v = mus_filt[off] + sRes[idx];
      sMu[idx] = v;
      out_mu[off] = v;
    }
    __syncthreads();
  }
}

// ---------------------------------------------------------------------------
extern "C" void kernel_launch(void* const* d_in, const int* in_sizes, int n_in,
                              void* d_out, int out_size, void* d_ws, size_t ws_size,
                              hipStream_t stream) {
  const float* A      = (const float*)d_in[0];
  const float* W      = (const float*)d_in[1];
  const float* Q      = (const float*)d_in[2];
  const float* R      = (const float*)d_in[3];
  const float* Sigma0 = (const float*)d_in[4];
  const float* mu0    = (const float*)d_in[5];
  const float* x_hat  = (const float*)d_in[6];

  float* out = (float*)d_out;
  float* out_mu   = out;                                  // [T,NMC,NT,64]
  float* out_chol = out + (size_t)T_LEN * NMC * NT * 64;  // [T,NT,64,64]

  float* ws = (float*)d_ws;
  const size_t mat = 4096;
  float* Sf_hist  = ws;  ws += (size_t)T_LEN * NT * mat;
  float* Sd_hist  = ws;  ws += (size_t)(T_LEN - 1) * NT * mat;
  float* L_hist   = ws;  ws += (size_t)(T_LEN - 1) * NT * mat;
  float* Cs_ws    = ws;  ws += (size_t)(T_LEN - 1) * NT * mat;
  float* md_hist  = ws;  ws += (size_t)(T_LEN - 1) * NMC * NT * 64;
  float* mus_filt = ws;  ws += (size_t)T_LEN * NMC * NT * 64;
  float* xt       = ws;

  xhat_transpose<<<2048, 256, 0, stream>>>(x_hat, xt);
  kf_filter<<<NT, 256, 0, stream>>>(A, W, Q, R, Sigma0, mu0, xt,
                                    Sf_hist, Sd_hist, L_hist, md_hist, mus_filt);
  kf_smooth_cov<<<T_LEN * NT, 256, 0, stream>>>(A, Sf_hist, Sd_hist, L_hist,
                                                Cs_ws, out_chol);
  kf_smooth_mean<<<NT, 256, 0, stream>>>(Cs_ws, md_hist, mus_filt, out_mu);
}